// MultimodalSequenceBuilder_76063870812687
// MI455X (gfx1250) — compile-verified
//
#include <hip/hip_runtime.h>

typedef __attribute__((ext_vector_type(2))) float v2f;
typedef __attribute__((ext_vector_type(4))) float f4;
typedef __attribute__((ext_vector_type(8))) float v8f;
typedef __attribute__((ext_vector_type(4))) int   v4i;

#define TOK 256
#define SEQ 1105
#define NDEMO 16
#define BATCH 256
#define NOBJ 64
#define KSLAB 16

// ---------------------------------------------------------------------------
// CDNA5 async global->LDS path (ASYNCcnt-tracked, no VGPR round trip).
// Builtin prototypes are typed-pointer form:
//   b32 : (int  as1*, int  as3*, imm offset, imm cpol)
//   b128: (int4 as1*, int4 as3*, imm offset, imm cpol)
// ---------------------------------------------------------------------------
#if __has_builtin(__builtin_amdgcn_global_load_async_to_lds_b128) && \
    __has_builtin(__builtin_amdgcn_global_load_async_to_lds_b32)
#define HAVE_ASYNC_LDS 1
#else
#define HAVE_ASYNC_LDS 0
#endif

typedef __attribute__((address_space(1))) int   as1_i32;
typedef __attribute__((address_space(3))) int   as3_i32;
typedef __attribute__((address_space(1))) v4i   as1_v4i;
typedef __attribute__((address_space(3))) v4i   as3_v4i;

__device__ __forceinline__ void cp_async_b128(void* lds, const void* g) {
#if HAVE_ASYNC_LDS
    __builtin_amdgcn_global_load_async_to_lds_b128(
        (as1_v4i*)(uintptr_t)g,
        (as3_v4i*)(uint32_t)(uintptr_t)lds,   // low 32 bits = LDS offset
        0, 0);
#else
    *(f4*)lds = *(const f4*)g;
#endif
}
__device__ __forceinline__ void cp_async_b32(void* lds, const void* g) {
#if HAVE_ASYNC_LDS
    __builtin_amdgcn_global_load_async_to_lds_b32(
        (as1_i32*)(uintptr_t)g,
        (as3_i32*)(uint32_t)(uintptr_t)lds,
        0, 0);
#else
    *(float*)lds = *(const float*)g;
#endif
}
__device__ __forceinline__ void wait_async() {
#if HAVE_ASYNC_LDS
#if __has_builtin(__builtin_amdgcn_s_wait_asynccnt)
    __builtin_amdgcn_s_wait_asynccnt(0);
#else
    asm volatile("s_wait_asynccnt 0x0" ::: "memory");
#endif
#endif
}

// ---------------------------------------------------------------------------
// 16-row x 256-col fp32 GEMM tile via V_WMMA_F32_16X16X4_F32, double-buffered
// LDS slabs of K=16 filled by async global->LDS loads.
// mode 0: object projection. Rows = [demo (16*256*64) | current (256*64)],
//         K = 256. Each 16-row tile -> 16 contiguous sequence slots of one b.
// mode 1: instruction projection. Rows = batch index, K = 768, s = 0.
// ---------------------------------------------------------------------------
__global__ __launch_bounds__(256)
void gemm16_wmma_kernel(const float* __restrict__ X1, const float* __restrict__ X2,
                        long rows1, int K,
                        const float* __restrict__ W, const float* __restrict__ bias,
                        float* __restrict__ out, int mode)
{
    __shared__ float Xs[2][16][20];      // 16 x KSLAB, pad->stride 20: A reads conflict-free
    __shared__ float Ws[2][KSLAB][260];  // KSLAB x 256, pad->stride 260: B reads conflict-free

    const int t     = threadIdx.x;
    const int lane  = t & 31;
    const int wave  = t >> 5;
    const int m     = lane & 15;   // A row / C column-lane
    const int khalf = lane >> 4;   // K-pair select (A/B); M+8 half (C/D)

    const long r0 = (long)blockIdx.x * 16;
    const float* Xrow0 = (r0 < rows1) ? (X1 + r0 * K) : (X2 + (r0 - rows1) * K);

    // tile -> output row base (sequence rows) + stride between tile rows
    long outRow0, outRowStride;
    if (mode == 0) {
        long b, s0;
        if (r0 < rows1) {
            long d   = r0 >> 14;          // / (256*64)
            long rem = r0 & 16383;
            b  = rem >> 6;
            s0 = 1 + d * 65 + (rem & 63);
        } else {
            long rc = r0 - rows1;
            b  = rc >> 6;
            s0 = 1 + (long)NDEMO * 65 + (rc & 63);
        }
        outRow0 = b * SEQ + s0;
        outRowStride = 1;
    } else {
        outRow0 = r0 * SEQ;
        outRowStride = SEQ;
    }

    v8f acc0 = {}; v8f acc1 = {};
    const int n0 = wave * 32 + (lane & 15);
    const int n1 = n0 + 16;

    const int nslabs = K >> 4;

    // per-thread staging coordinates
    const int xrow = t >> 4;             // 0..15
    const int xkl  = t & 15;             // 0..15

    // ---- issue slab 0 ----
    {
        cp_async_b32(&Xs[0][xrow][xkl], Xrow0 + (long)xrow * K + xkl);
        const float* wsrc = W;
        #pragma unroll
        for (int q = 0; q < 4; ++q) {
            const int flat = (t << 2) + (q << 10);
            cp_async_b128(&Ws[0][flat >> 8][flat & 255], wsrc + flat);
        }
    }
    wait_async();
    __syncthreads();

    for (int s = 0; s < nslabs; ++s) {
        const int buf = s & 1;

        // ---- issue slab s+1 into the other buffer (overlaps with compute) ----
        if (s + 1 < nslabs) {
            const int k0n = (s + 1) << 4;
            cp_async_b32(&Xs[buf ^ 1][xrow][xkl], Xrow0 + (long)xrow * K + k0n + xkl);
            const float* wsrc = W + (long)k0n * TOK;
            #pragma unroll
            for (int q = 0; q < 4; ++q) {
                const int flat = (t << 2) + (q << 10);
                cp_async_b128(&Ws[buf ^ 1][flat >> 8][flat & 255], wsrc + flat);
            }
            // pull slab s+2 toward L2 while HBM streams
            if (s + 2 < nslabs)
                __builtin_prefetch(W + (long)((s + 2) << 4) * TOK + (t << 2), 0, 0);
        }

        // ---- compute: 4 K-steps of 4 -> 8 WMMA per slab per wave ----
        #pragma unroll
        for (int i = 0; i < 4; ++i) {
            const int kk = (i << 2) + (khalf << 1);
            v2f a = *(const v2f*)&Xs[buf][m][kk];                 // ds_load_b64
            v2f b0v = { Ws[buf][kk][n0], Ws[buf][kk + 1][n0] };
            v2f b1v = { Ws[buf][kk][n1], Ws[buf][kk + 1][n1] };
            acc0 = __builtin_amdgcn_wmma_f32_16x16x4_f32(
                false, a, false, b0v, (short)0, acc0, false, false);
            acc1 = __builtin_amdgcn_wmma_f32_16x16x4_f32(
                false, a, false, b1v, (short)0, acc1, false, false);
        }

        wait_async();      // slab s+1 resident in LDS
        __syncthreads();   // all waves done reading buf before it is recycled
    }

    // ---- bias + store: VGPR v holds (M = v + 8*khalf, N = lane&15) ----
    const float bv0 = bias[n0];
    const float bv1 = bias[n1];
    #pragma unroll
    for (int v = 0; v < 8; ++v) {
        const int rowm = v + (khalf << 3);
        float* orow = out + (outRow0 + (long)rowm * outRowStride) * TOK;
        orow[n0] = acc0[v] + bv0;
        orow[n1] = acc1[v] + bv1;
    }
}

// ---------------------------------------------------------------------------
// Action projection: (16,256,7) @ (7,256) + b. K=7 -> plain VALU, negligible.
// ---------------------------------------------------------------------------
__global__ __launch_bounds__(256)
void act_proj_kernel(const float* __restrict__ A, const float* __restrict__ W,
                     const float* __restrict__ bias, float* __restrict__ out)
{
    const int r    = blockIdx.x;      // d*256 + b
    const int tcol = threadIdx.x;
    const int d = r >> 8;
    const int b = r & 255;
    const float* a = A + (long)r * 7;
    float sum = bias[tcol];
    #pragma unroll
    for (int f = 0; f < 7; ++f)
        sum += a[f] * W[f * TOK + tcol];
    const long s = 1 + (long)d * 65 + 64;
    out[((long)b * SEQ + s) * TOK + tcol] = sum;
}

extern "C" void kernel_launch(void* const* d_in, const int* in_sizes, int n_in,
                              void* d_out, int out_size, void* d_ws, size_t ws_size,
                              hipStream_t stream)
{
    (void)in_sizes; (void)n_in; (void)out_size; (void)d_ws; (void)ws_size;

    const float* instr    = (const float*)d_in[0];  // (256, 768)
    const float* demo_obj = (const float*)d_in[1];  // (16, 256, 64, 256)
    const float* demo_act = (const float*)d_in[2];  // (16, 256, 7)
    const float* cur_obj  = (const float*)d_in[3];  // (256, 64, 256)
    const float* W_instr  = (const float*)d_in[4];  // (768, 256)
    const float* b_instr  = (const float*)d_in[5];  // (256)
    const float* W_act    = (const float*)d_in[6];  // (7, 256)
    const float* b_act    = (const float*)d_in[7];  // (256)
    const float* W_obj    = (const float*)d_in[8];  // (256, 256)
    const float* b_obj    = (const float*)d_in[9];  // (256)
    float* out = (float*)d_out;                     // (256, 1105, 256)

    const long demoRows = (long)NDEMO * BATCH * NOBJ;     // 262144
    const long objRows  = demoRows + (long)BATCH * NOBJ;  // 278528
    const unsigned objTiles = (unsigned)(objRows / 16);   // 17408

    gemm16_wmma_kernel<<<dim3(objTiles), dim3(256), 0, stream>>>(
        demo_obj, cur_obj, demoRows, 256, W_obj, b_obj, out, 0);

    gemm16_wmma_kernel<<<dim3(16), dim3(256), 0, stream>>>(
        instr, instr, (long)BATCH, 768, W_instr, b_instr, out, 1);

    act_proj_kernel<<<dim3(4096), dim3(256), 0, stream>>>(
        demo_act, W_act, b_act, out);
}